// SPDConv2D_5901285065060
// MI455X (gfx1250) — compile-verified
//
#include <hip/hip_runtime.h>
#include <hip/hip_bf16.h>
#include <math.h>
#include <stdint.h>

// ---------------- problem constants (from reference) ----------------
#define BB   4
#define CC   8
#define HH   20
#define WW   20
#define NN   8           // SPD matrix dim
#define OO   16          // output channels
#define KERN 3
#define STRIDE 2
#define HO   9
#define WO   9
#define NWIN (KERN*KERN*CC)   // 72
#define NSWEEP 6
#define EPS  1e-6f

typedef float v2f __attribute__((ext_vector_type(2)));
typedef float v8f __attribute__((ext_vector_type(8)));
typedef unsigned int v4u __attribute__((ext_vector_type(4)));
typedef unsigned int v8u __attribute__((ext_vector_type(8)));

#define WSYNC() __builtin_amdgcn_wave_barrier()

// Per-wave scratch layout (in units of floats), 8 slots of 128 (pair of 8x8):
//   M=0, L=128, LI=256, Z=384, V=512, T1=640, T2=768, DUMP=896
#define SM   0
#define SL   128
#define SLI  256
#define SZ   384
#define SV   512
#define ST1  640
#define ST2  768
#define SDMP 896

// ------------------------------------------------------------------
// Batched 8x8 x 8x8 fp32 matmul for TWO independent problems per wave
// using V_WMMA_F32_16X16X4_F32.  A stacked [Aa;Ab] (16x8) x B wide
// [Ba|Bb] (8x16): diagonal 8x8 blocks of the 16x16 result are Aa*Ba and
// Ab*Bb.  K=8 -> two wmma issues.  Branchless write-back (dump slot).
// ------------------------------------------------------------------
__device__ __forceinline__ void mm8_pair(const float* __restrict__ Aa,
                                         const float* __restrict__ Ab,
                                         const float* __restrict__ Ba,
                                         const float* __restrict__ Bb,
                                         float* __restrict__ Ra,
                                         float* __restrict__ Rb,
                                         float* __restrict__ dmp,
                                         int lane, bool transB) {
    const int half = lane >> 4;        // 0: lanes 0-15, 1: lanes 16-31
    const int lm   = lane & 15;        // M index (A) / N index (B)
    const float* Asrc = (lm < 8) ? Aa : Ab;
    const float* Bsrc = (lm < 8) ? Ba : Bb;
    const int am = lm & 7;
    const int bn = lm & 7;

    v8f c = {};
#pragma unroll
    for (int k0 = 0; k0 < 8; k0 += 4) {
        const int ka = k0 + 2 * half;          // ISA A 16x4 layout
        v2f a, b;
        a.x = Asrc[am * 8 + ka];
        a.y = Asrc[am * 8 + ka + 1];
        if (transB) {
            b.x = Bsrc[bn * 8 + ka];
            b.y = Bsrc[bn * 8 + ka + 1];
        } else {
            b.x = Bsrc[ka * 8 + bn];
            b.y = Bsrc[(ka + 1) * 8 + bn];
        }
        c = __builtin_amdgcn_wmma_f32_16x16x4_f32(
                false, a, false, b, (short)0, c, false, false);
    }
    // branchless write-back of the two diagonal 8x8 blocks
    const bool wr = (half == 0) ? (lm < 8) : (lm >= 8);
    float* R = (half == 0) ? Ra : Rb;
    const int col = lm & 7;
#pragma unroll
    for (int r = 0; r < 8; ++r) {
        float* p = wr ? &R[r * 8 + col] : dmp;
        *p = c[r];
    }
    WSYNC();
}

// Cooperative Cholesky (pair): lanes 0-7 rows of a, 16-23 rows of b.
// Fully uniform control flow; predication via dump-address selects.
__device__ __forceinline__ void chol8(float* S, int lane) {
    const int  h    = (lane < 16) ? 0 : 64;
    const int  i    = lane & 7;
    const bool act  = ((lane & 15) < 8);
    const int  dump = SDMP + lane;
#pragma unroll
    for (int j = 0; j < 8; ++j)
        S[act ? (SL + h + i * 8 + j) : dump] = 0.f;
    WSYNC();
#pragma unroll
    for (int j = 0; j < 8; ++j) {
        float s = S[SM + h + i * 8 + j];
#pragma unroll
        for (int k = 0; k < j; ++k)
            s -= S[SL + h + i * 8 + k] * S[SL + h + j * 8 + k];
        const float dv = sqrtf(fmaxf(s, 1e-20f));
        S[(act && i == j) ? (SL + h + j * 8 + j) : dump] = dv;
        WSYNC();
        const float d = S[SL + h + j * 8 + j];
        S[(act && i > j) ? (SL + h + i * 8 + j) : dump] = s / d;
        WSYNC();
    }
}

// Lower-triangular inverse (pair): lane j owns column j (no cross-lane deps).
__device__ __forceinline__ void trinv8(float* S, int lane) {
    const int  h    = (lane < 16) ? 0 : 64;
    const int  j    = lane & 7;
    const bool act  = ((lane & 15) < 8);
    const int  dump = SDMP + lane;
#pragma unroll
    for (int i = 0; i < 8; ++i)
        S[act ? (SLI + h + i * 8 + j) : dump] = 0.f;
    S[act ? (SLI + h + j * 8 + j) : dump] = 1.f / S[SL + h + j * 8 + j];
#pragma unroll
    for (int i = 1; i < 8; ++i) {
        float s = 0.f;
#pragma unroll
        for (int k = 0; k < i; ++k)            // R[k][j]==0 for k<j
            s += S[SL + h + i * 8 + k] * S[SLI + h + k * 8 + j];
        const float v = -s / S[SL + h + i * 8 + i];
        S[(act && i > j) ? (SLI + h + i * 8 + j) : dump] = v;
    }
    WSYNC();
}

// In-place symmetrize (pair) of slot `off`: lane j owns column j.
__device__ __forceinline__ void sym8(float* S, int off, int lane) {
    const int  h    = (lane < 16) ? 0 : 64;
    const int  j    = lane & 7;
    const bool act  = ((lane & 15) < 8);
    const int  dump = SDMP + lane;
    float col[8];
#pragma unroll
    for (int i = 0; i < 8; ++i)
        col[i] = 0.5f * (S[off + h + i * 8 + j] + S[off + h + j * 8 + i]);
    WSYNC();
#pragma unroll
    for (int i = 0; i < 8; ++i)
        S[act ? (off + h + i * 8 + j) : dump] = col[i];
    WSYNC();
}

// Cyclic Jacobi eigensolver (pair), branchless rotations.
__device__ __forceinline__ void jacobi8(float* S, int lane) {
    const int  h    = (lane < 16) ? 0 : 64;
    const int  j    = lane & 7;
    const bool act  = ((lane & 15) < 8);
    const int  dump = SDMP + lane;
#pragma unroll
    for (int i = 0; i < 8; ++i)
        S[act ? (SV + h + i * 8 + j) : dump] = (i == j) ? 1.f : 0.f;
    WSYNC();
    for (int sweep = 0; sweep < NSWEEP; ++sweep) {
        for (int p = 0; p < 7; ++p) {
            for (int q = p + 1; q < 8; ++q) {
                const float app = S[SZ + h + p * 8 + p];
                const float aqq = S[SZ + h + q * 8 + q];
                const float apq = S[SZ + h + p * 8 + q];
                const bool  skip =
                    fabsf(apq) < 1e-14f * (fabsf(app) + fabsf(aqq)) + 1e-30f;
                const float tau = (aqq - app) / (2.f * apq);
                const float t   = copysignf(1.f, tau) /
                                  (fabsf(tau) + sqrtf(1.f + tau * tau));
                float c = 1.f / sqrtf(1.f + t * t);
                float s = t * c;
                c = skip ? 1.f : c;
                s = skip ? 0.f : s;
                // rotate rows p,q (lane j owns column j)
                {
                    const float zp = S[SZ + h + p * 8 + j];
                    const float zq = S[SZ + h + q * 8 + j];
                    S[act ? (SZ + h + p * 8 + j) : dump] = c * zp - s * zq;
                    S[act ? (SZ + h + q * 8 + j) : dump] = s * zp + c * zq;
                }
                WSYNC();
                // rotate cols p,q (lane j owns row j); V columns too
                {
                    const float zp = S[SZ + h + j * 8 + p];
                    const float zq = S[SZ + h + j * 8 + q];
                    S[act ? (SZ + h + j * 8 + p) : dump] = c * zp - s * zq;
                    S[act ? (SZ + h + j * 8 + q) : dump] = s * zp + c * zq;
                    const float vp = S[SV + h + j * 8 + p];
                    const float vq = S[SV + h + j * 8 + q];
                    S[act ? (SV + h + j * 8 + p) : dump] = c * vp - s * vq;
                    S[act ? (SV + h + j * 8 + q) : dump] = s * vp + c * vq;
                }
                WSYNC();
            }
        }
    }
}

// ------------------------------------------------------------------
// Kernel 1: t[o][j] = wn[o][j] / cumsum(wn[o][:j+1]),  wn = w^2/(sum+eps)
// ------------------------------------------------------------------
__global__ void spd_tnorm_kernel(const float* __restrict__ w,
                                 float* __restrict__ t) {
    const int o = threadIdx.x;
    if (o >= OO) return;
    float Ssum = 0.f;
    for (int j = 0; j < NWIN; ++j) { const float x = w[o * NWIN + j]; Ssum += x * x; }
    const float inv = 1.f / (Ssum + EPS);
    float cum = 0.f;
    for (int j = 0; j < NWIN; ++j) {
        const float x  = w[o * NWIN + j];
        const float wn = x * x * inv;
        cum += wn;
        t[o * NWIN + j] = wn / cum;
    }
}

// ------------------------------------------------------------------
// Kernel 2: main recursive Fréchet-mean scan.
// Block = one site (b,ho,wo); 8 waves; wave w handles oc pair (2w,2w+1).
// Window staged by ONE Tensor-Data-Mover descriptor (3D tile:
//   dim0 = KERN*64 contiguous, dim1 = ki (stride W*64), dim2 = c
//   (stride H*W*64)) -> LDS layout [c][ki][kj][64].
// ------------------------------------------------------------------
__global__ void __launch_bounds__(256)
spd_fm_kernel(const float* __restrict__ xg,
              const float* __restrict__ tt,
              float* __restrict__ outg) {
    __shared__ float Xs[NWIN * 64];          // 18 KB: full window
    __shared__ float Wk[8 * 8 * 128];        // 32 KB: per-wave pair scratch

    const int site = blockIdx.x;
    const int wo   = site % WO;
    const int tmp  = site / WO;
    const int ho   = tmp % HO;
    const int b    = tmp / HO;

    const int wave = threadIdx.x >> 5;
    const int lane = threadIdx.x & 31;

    // ---- TDM: async tensor load of the whole window into LDS ----
    if (wave == 0) {
        const uint64_t gbase = (uint64_t)(uintptr_t)
            (xg + ((((size_t)b * CC + 0) * HH + (ho * STRIDE)) * WW +
                   (wo * STRIDE)) * 64);
        const uint32_t ldsa = (uint32_t)(uintptr_t)&Xs[0];

        const uint32_t TD0 = WW * 64;         // tensor dim0 length (elements)
        const uint32_t TD1 = HH;              // tensor dim1 length (lines)
        const uint32_t TD2 = CC;              // tensor dim2 length
        const uint32_t TL0 = KERN * 64;       // tile dim0 = 192 floats
        const uint32_t TL1 = KERN;            // tile dim1 = 3 (ki)
        const uint32_t TL2 = CC;              // tile dim2 = 8 (c)
        const uint32_t S0  = WW * 64;         // dim0 stride = 1280
        const uint32_t S1  = HH * WW * 64;    // dim1 stride = 25600

        v4u g0;
        g0[0] = 1u;                                    // count=1, user D#
        g0[1] = ldsa;                                  // lds_addr
        g0[2] = (uint32_t)gbase;                       // global_addr[31:0]
        g0[3] = (uint32_t)(gbase >> 32) | (2u << 30);  // addr[56:32] | type=2

        v8u g1;
        g1[0] = (2u << 16);                            // data_size = 4B
        g1[1] = (TD0 & 0xFFFFu) << 16;                 // tensor_dim0[15:0]
        g1[2] = (TD0 >> 16) | ((TD1 & 0xFFFFu) << 16); // td0 hi | td1 lo
        g1[3] = (TD1 >> 16) | (TL0 << 16);             // td1 hi | tile_dim0
        g1[4] = TL1 | (TL2 << 16);                     // tile_dim1 | tile_dim2
        g1[5] = S0;                                    // dim0_stride[31:0]
        g1[6] = (S1 & 0xFFFFu) << 16;                  // s0 hi16=0 | s1 lo16
        g1[7] = S1 >> 16;                              // s1 hi32

        v4u g2;
        g2[0] = TD2;                                   // tensor_dim2
        g2[1] = 1u;                                    // tensor_dim3 (unused)
        g2[2] = 0u;                                    // dim2_stride lo (unused)
        g2[3] = 0u;                                    // dim2_stride hi | tile_dim3=0
        v4u g3;
        g3[0] = 0u; g3[1] = 0u; g3[2] = 0u; g3[3] = 0u;

        asm volatile("tensor_load_to_lds %0, %1, %2, %3"
                     :: "s"(g0), "s"(g1), "s"(g2), "s"(g3)
                     : "memory");
        __builtin_amdgcn_s_wait_tensorcnt(0);
    }
    __syncthreads();

    float* S   = &Wk[wave * 8 * 128];
    float* dmp = &S[SDMP + lane];
    const int o0 = wave * 2, o1 = o0 + 1;

    // M = I (both channels), branchless
    {
        const int  h    = (lane < 16) ? 0 : 64;
        const int  j    = lane & 7;
        const bool act  = ((lane & 15) < 8);
        const int  dump = SDMP + lane;
#pragma unroll
        for (int i = 0; i < 8; ++i)
            S[act ? (SM + h + i * 8 + j) : dump] = (i == j) ? 1.f : 0.f;
    }
    WSYNC();

    for (int k = 0; k < NWIN; ++k) {
        // reference window order j=(ki*3+kj)*C+c -> TDM LDS slot (c*3+ki)*3+kj
        const int c  = k % CC;
        const int kk = k / CC;
        const int kj = kk % KERN;
        const int ki = kk / KERN;
        const float* Xk = &Xs[((c * KERN + ki) * KERN + kj) * 64];

        chol8 (S, lane);                                      // L  = chol(M)
        trinv8(S, lane);                                      // LI = L^-1
        mm8_pair(S+SLI, S+SLI+64, Xk,    Xk,       S+ST1, S+ST1+64, dmp, lane, false);
        mm8_pair(S+ST1, S+ST1+64, S+SLI, S+SLI+64, S+SZ,  S+SZ +64, dmp, lane, true);
        sym8(S, SZ, lane);
        jacobi8(S, lane);                                     // eig(Z) -> V, diag

        // T1 := V * diag(ev^t)
        {
            const float tk  = (lane < 16) ? tt[o0 * NWIN + k] : tt[o1 * NWIN + k];
            const int   h   = (lane < 16) ? 0 : 64;
            const bool  act = ((lane & 15) < 8);
            const int   e   = lane & 7;
            const int   dump = SDMP + lane;
            const float ev  = fmaxf(S[SZ + h + e * 8 + e], 1e-10f);
            const float evt = exp2f(tk * log2f(ev));
#pragma unroll
            for (int i = 0; i < 8; ++i)
                S[act ? (ST1 + h + i * 8 + e) : dump] =
                    S[SV + h + i * 8 + e] * evt;
        }
        WSYNC();

        mm8_pair(S+ST1, S+ST1+64, S+SV,  S+SV +64, S+ST2, S+ST2+64, dmp, lane, true);  // Zt
        mm8_pair(S+SL,  S+SL +64, S+ST2, S+ST2+64, S+ST1, S+ST1+64, dmp, lane, false); // L*Zt
        mm8_pair(S+ST1, S+ST1+64, S+SL,  S+SL +64, S+SM,  S+SM +64, dmp, lane, true);  // *L^T
        sym8(S, SM, lane);
    }

    // ---- write out: out[b][o][ho][wo][8][8] ----
    const size_t base0 = ((((size_t)b * OO + o0) * HO + ho) * WO + wo) * 64;
    const size_t base1 = ((((size_t)b * OO + o1) * HO + ho) * WO + wo) * 64;
    for (int e = lane; e < 64; e += 32) {
        outg[base0 + e] = S[SM + e];
        outg[base1 + e] = S[SM + 64 + e];
    }
}

// ------------------------------------------------------------------
extern "C" void kernel_launch(void* const* d_in, const int* in_sizes, int n_in,
                              void* d_out, int out_size, void* d_ws, size_t ws_size,
                              hipStream_t stream) {
    const float* x  = (const float*)d_in[0];   // (4,8,20,20,8,8) f32
    const float* wm = (const float*)d_in[1];   // (16,72) f32
    float* out = (float*)d_out;                // (4,16,9,9,8,8) f32
    float* tt  = (float*)d_ws;                 // 16*72 f32 scratch

    spd_tnorm_kernel<<<1, 32, 0, stream>>>(wm, tt);
    spd_fm_kernel<<<BB * HO * WO, 256, 0, stream>>>(x, tt, out);
}